// PatchContrastiveLoss_60413009985852
// MI455X (gfx1250) — compile-verified
//
#include <hip/hip_runtime.h>
#include <math.h>

// Static config mirroring the reference
#define N_     2
#define SF_    2
#define KP_    8
#define C_     3
#define H_     64
#define W_     64
#define NBINS_ 256
#define P_     (H_ * W_)       // 4096 pixels
#define D_     (C_ * NBINS_)   // 768 descriptor dim
#define ROWS_  (KP_ * SF_)     // 16 rows per image in the sim matrix
#define CHUNKS_ 4              // pixel chunks per (b,c) block
#define CHUNK_P_ (P_ / CHUNKS_)  // 1024 pixels per chunk

typedef __attribute__((ext_vector_type(2))) float v2f;
typedef __attribute__((ext_vector_type(8))) float v8f;

// ---------------------------------------------------------------------------
// Kernel 1: Gaussian mask * image -> soft histogram (256 bins), bin0 zeroed.
// One block per (b, c) pair: 96 blocks x 1024 threads (32 wave32 waves/WGP).
// Thread t = (chunk = t>>8, bin = t&255): each thread reduces 1024 pixels for
// its bin; 4 chunk-partials per bin are combined with a fixed-order LDS sum
// (deterministic, no float atomics). Lanes in a wave share a chunk, so the
// inner LDS read is a broadcast (no bank conflicts).
// Output layout: hist[(n*16 + kp*2 + sf)*768 + c*256 + bin]  (the reference's
// post-transpose (N,KP,SF,C*256) row layout, so downstream is a plain GEMM).
// ---------------------------------------------------------------------------
__global__ __launch_bounds__(1024) void pcl_hist_kernel(
    const float* __restrict__ coords,   // (N,SF,KP,2)
    const float* __restrict__ images,   // (N,SF,C,H,W)
    float* __restrict__ hist)           // (N*16, 768)
{
    __shared__ float s_pv[P_];          // 16 KB: image*mask patch values
    __shared__ float s_red[1024];       // 4 KB: chunk-partial histograms

    const int blk = blockIdx.x;         // 0..95 = b*C + c
    const int b   = blk / C_;           // (n*SF+sf)*KP + kp
    const int c   = blk % C_;
    const int t   = threadIdx.x;

    const int kp  = b % KP_;
    const int nsf = b / KP_;            // n*SF + sf
    const int sf  = nsf % SF_;
    const int n   = nsf / SF_;

    const float cx = coords[b * 2 + 0] * (float)W_;
    const float cy = coords[b * 2 + 1] * (float)H_;
    const float inv2s2 = 1.0f / (2.0f * 6.0f * 6.0f);   // 1/(2*STD^2)
    const int   imgbase = nsf * (C_ * P_) + c * P_;

    // gfx1250 global_prefetch_b8: pull the 16KB image tile toward the WGP
    __builtin_prefetch(images + imgbase + t * 4, 0, 0);

    // Stage patch values (image * per-keypoint Gaussian) into LDS.
    for (int p = t; p < P_; p += 1024) {
        float y  = (float)(p >> 6);
        float x  = (float)(p & 63);
        float dy = y - cy, dx = x - cx;
        float m  = __expf(-(dy * dy + dx * dx) * inv2s2);
        s_pv[p]  = images[imgbase + p] * m;
    }
    __syncthreads();

    const int   bin    = t & 255;
    const int   chunk  = t >> 8;        // 0..3 (uniform per wave)
    const float center = ((float)bin + 0.5f) / (float)NBINS_;
    const float hd     = 0.5f / (float)NBINS_;          // delta/2
    const float invBW  = 1000.0f;                        // 1/bandwidth

    float acc = 0.0f;
    const int p0 = chunk * CHUNK_P_;
    #pragma unroll 8
    for (int p = p0; p < p0 + CHUNK_P_; ++p) {
        float d  = s_pv[p] - center;               // LDS broadcast read
        float zp = (d + hd) * invBW;
        float zm = (d - hd) * invBW;
        float sp = 1.0f / (1.0f + __expf(-zp));
        float sm = 1.0f / (1.0f + __expf(-zm));
        acc += sp - sm;
    }
    s_red[t] = acc;
    __syncthreads();

    if (t < NBINS_) {
        // fixed summation order -> deterministic
        float v = ((s_red[t] + s_red[t + 256]) + (s_red[t + 512] + s_red[t + 768]));
        const int row = (n * ROWS_ + kp * SF_ + sf) * D_;
        hist[row + c * NBINS_ + t] = (t == 0) ? 0.0f : v;   // kill bin 0
    }
}

// ---------------------------------------------------------------------------
// Kernel 2: L2-normalize each 768-dim row (32 rows). 256 threads, 3 elems each.
// ---------------------------------------------------------------------------
__global__ __launch_bounds__(256) void pcl_norm_kernel(
    const float* __restrict__ hist, float* __restrict__ normed)
{
    __shared__ float red[256];
    const int rr = blockIdx.x;     // 0..31
    const int t  = threadIdx.x;
    const float* src = hist + rr * D_;

    float v0 = src[t], v1 = src[t + 256], v2 = src[t + 512];
    red[t] = v0 * v0 + v1 * v1 + v2 * v2;
    __syncthreads();
    for (int s = 128; s > 0; s >>= 1) {
        if (t < s) red[t] += red[t + s];
        __syncthreads();
    }
    float inv = 1.0f / (sqrtf(red[0]) + 1e-8f);
    normed[rr * D_ + t]       = v0 * inv;
    normed[rr * D_ + t + 256] = v1 * inv;
    normed[rr * D_ + t + 512] = v2 * inv;
}

// ---------------------------------------------------------------------------
// Kernel 3: per-image Gram matrix sim = R * R^T (16x16x768) via
// V_WMMA_F32_16X16X4_F32 (f32 precision), then masked contrastive reduction.
// One wave per image n (wave32); both A and B operands are the same register
// pair (Gram-matrix symmetry of the 16x4 / 4x16 f32 WMMA layouts).
// ---------------------------------------------------------------------------
__global__ __launch_bounds__(64) void pcl_simloss_kernel(
    const float* __restrict__ normed, float* __restrict__ out)
{
    __shared__ float s_pos[2], s_neg[2];
    const int lane = threadIdx.x & 31;
    const int wave = threadIdx.x >> 5;          // image index n
    const float* R = normed + wave * ROWS_ * D_;

    // A-tile addressing: lanes 0-15 -> K {0,1}, lanes 16-31 -> K {2,3}
    const int m     = lane & 15;
    const int khalf = (lane >> 4) * 2;

    v8f acc = {};
    for (int k = 0; k < D_; k += 4) {
        v2f a = *(const v2f*)(R + m * D_ + k + khalf);
        // (neg_a, A, neg_b, B, c_mod, C, reuse_a, reuse_b)
        acc = __builtin_amdgcn_wmma_f32_16x16x4_f32(
            false, a, false, a, (short)0, acc, false, false);
    }

    // acc[r] = sim[i][j], i = r + (lane>=16 ? 8 : 0), j = lane&15
    const int j     = lane & 15;
    const int ibase = (lane >> 4) * 8;
    float pos = 0.0f, neg = 0.0f;
    #pragma unroll
    for (int r = 0; r < 8; ++r) {
        int   i = ibase + r;
        float s = acc[r];
        bool same_kp = (i >> 1) == (j >> 1);
        bool same_sf = (i & 1) == (j & 1);
        if (same_kp && !same_sf) pos += 1.0f - s;
        if (!same_kp)            neg += fmaxf(s - 0.5f, 0.0f);
    }
    // wave32 reduction
    for (int off = 16; off > 0; off >>= 1) {
        pos += __shfl_xor(pos, off, 32);
        neg += __shfl_xor(neg, off, 32);
    }
    if (lane == 0) { s_pos[wave] = pos; s_neg[wave] = neg; }
    __syncthreads();
    if (threadIdx.x == 0) {
        // n_pos = 16*N = 32, n_neg = 224*N = 448
        out[0] = (s_pos[0] + s_pos[1]) / 32.0f +
                 (s_neg[0] + s_neg[1]) / 448.0f;
    }
}

extern "C" void kernel_launch(void* const* d_in, const int* in_sizes, int n_in,
                              void* d_out, int out_size, void* d_ws, size_t ws_size,
                              hipStream_t stream) {
    (void)in_sizes; (void)n_in; (void)out_size; (void)ws_size;
    const float* coords = (const float*)d_in[0];   // 64 floats
    const float* images = (const float*)d_in[1];   // 49152 floats
    float* out    = (float*)d_out;                 // scalar loss
    float* hist   = (float*)d_ws;                  // 32*768 floats
    float* normed = hist + (N_ * ROWS_ * D_);      // 32*768 floats

    pcl_hist_kernel<<<N_ * SF_ * KP_ * C_, 1024, 0, stream>>>(coords, images, hist);
    pcl_norm_kernel<<<N_ * ROWS_, 256, 0, stream>>>(hist, normed);
    pcl_simloss_kernel<<<1, 64, 0, stream>>>(normed, out);
}